// DTCWTForward_19945828122832
// MI455X (gfx1250) — compile-verified
//
#include <hip/hip_runtime.h>

typedef float v2f __attribute__((ext_vector_type(2)));
typedef float v8f __attribute__((ext_vector_type(8)));

#define KT 24                    // padded K for Toeplitz WMMA (16 + 2*3 halo -> 24)
#define SQH 0.70710678118654752440f

__device__ __forceinline__ int refl(int c, int r) {
    // symmetric reflection (edge included); valid for -r <= c < 2r
    return (c < 0) ? (-1 - c) : ((c >= r) ? (2 * r - 1 - c) : c);
}

// ---------------------------------------------------------------------------
// Level-1 row filter (along W): lo = x * h0o (5-tap), hi = x * h1o (7-tap)
// One block per (row, image); row staged in LDS with reflected halo.
// ---------------------------------------------------------------------------
__global__ __launch_bounds__(256) void rowfilt_l1(
    const float* __restrict__ x, float* __restrict__ lo, float* __restrict__ hi,
    const float* __restrict__ h0o, const float* __restrict__ h1o)
{
    __shared__ float row[518];           // 512 + 2*3 halo
    const int y  = blockIdx.x;
    const int im = blockIdx.y;
    const float* xr = x + ((size_t)im * 512 + y) * 512;
    for (int idx = threadIdx.x; idx < 518; idx += 256) {
        row[idx] = xr[refl(idx - 3, 512)];
    }
    __syncthreads();
    float f0[5], f1[7];
    #pragma unroll
    for (int j = 0; j < 5; ++j) f0[j] = h0o[j];
    #pragma unroll
    for (int j = 0; j < 7; ++j) f1[j] = h1o[j];
    float* lor = lo + ((size_t)im * 512 + y) * 512;
    float* hir = hi + ((size_t)im * 512 + y) * 512;
    for (int p = threadIdx.x; p < 512; p += 256) {
        float accl = 0.f, acch = 0.f;
        #pragma unroll
        for (int j = 0; j < 5; ++j) accl += f0[4 - j] * row[p + 1 + j]; // m2=2, lds shift 3
        #pragma unroll
        for (int j = 0; j < 7; ++j) acch += f1[6 - j] * row[p + j];     // m2=3
        lor[p] = accl;
        hir[p] = acch;
    }
}

// ---------------------------------------------------------------------------
// Level-1 column filter via fp32 WMMA Toeplitz + fused q2c -> yh[0].
// One wave per 16x16 output tile. Out(16x16) = A_toeplitz(16xKT) x B_data(KTx16).
// ---------------------------------------------------------------------------
__global__ __launch_bounds__(32) void colfilt_l1_wmma(
    const float* __restrict__ lo, const float* __restrict__ hi,
    const float* __restrict__ h0o, const float* __restrict__ h1o,
    float* __restrict__ ll, float* __restrict__ yh0)
{
    __shared__ float A0[16 * KT], A1[16 * KT];     // Toeplitz for h0o / h1o
    __shared__ float Blo[16 * KT], Bhi[16 * KT];   // data, stored transposed [n][k]
    __shared__ float Tlh[256], Thl[256], Thh[256]; // band tiles for q2c
    const int lane = threadIdx.x;                  // 0..31
    const int w0 = blockIdx.x * 16;
    const int h0 = blockIdx.y * 16;
    const int im = blockIdx.z;
    const float* loim = lo + (size_t)im * (512 * 512);
    const float* hiim = hi + (size_t)im * (512 * 512);

    // Toeplitz: out[h0+i] = sum_j f[m-1-j] * In[h0+i-m2+j]; B row k = In[h0-3+k].
    // 5-tap: j = k-i-1 ; 7-tap: j = k-i.
    for (int idx = lane; idx < 16 * KT; idx += 32) {
        int i = idx / KT, k = idx % KT;
        int j0 = k - i - 1;
        int j1 = k - i;
        A0[idx] = (j0 >= 0 && j0 < 5) ? h0o[4 - j0] : 0.0f;
        A1[idx] = (j1 >= 0 && j1 < 7) ? h1o[6 - j1] : 0.0f;
    }
    // Data tiles: B[n][k] = In[refl(h0-3+k)][w0+n]  (coalesced 16-lane row reads)
    for (int idx = lane; idx < 16 * KT; idx += 32) {
        int k = idx >> 4, n = idx & 15;
        int r = refl(h0 - 3 + k, 512);
        Blo[n * KT + k] = loim[(size_t)r * 512 + w0 + n];
        Bhi[n * KT + k] = hiim[(size_t)r * 512 + w0 + n];
    }
    __syncthreads();

    v8f cll = {}, clh = {}, chl = {}, chh = {};
    const int idx16 = lane & 15;             // M for A, N for B
    const int koff  = (lane >> 4) << 1;      // lanes 16-31 hold K pair (k0+2,k0+3)
    #pragma unroll
    for (int kb = 0; kb < KT; kb += 4) {
        v2f a0 = *(const v2f*)&A0[idx16 * KT + kb + koff];
        v2f a1 = *(const v2f*)&A1[idx16 * KT + kb + koff];
        v2f bl = *(const v2f*)&Blo[idx16 * KT + kb + koff];
        v2f bh = *(const v2f*)&Bhi[idx16 * KT + kb + koff];
        cll = __builtin_amdgcn_wmma_f32_16x16x4_f32(false, a0, false, bl, (short)0, cll, false, false);
        clh = __builtin_amdgcn_wmma_f32_16x16x4_f32(false, a1, false, bl, (short)0, clh, false, false);
        chl = __builtin_amdgcn_wmma_f32_16x16x4_f32(false, a0, false, bh, (short)0, chl, false, false);
        chh = __builtin_amdgcn_wmma_f32_16x16x4_f32(false, a1, false, bh, (short)0, chh, false, false);
    }

    // D layout: lane holds column n=lane&15; VGPR v holds row v (+8 for lanes>=16).
    const int n = lane & 15;
    const int mbase = (lane >> 4) << 3;
    float* llim = ll + (size_t)im * (512 * 512);
    #pragma unroll
    for (int v = 0; v < 8; ++v) {
        int m = mbase + v;
        llim[(size_t)(h0 + m) * 512 + w0 + n] = cll[v];
        Tlh[m * 16 + n] = clh[v];
        Thl[m * 16 + n] = chl[v];
        Thh[m * 16 + n] = chh[v];
    }
    __syncthreads();

    // q2c + band stacking: yh0[im][band][i][j][reim], H2=W2=256
    const size_t bs = (size_t)256 * 256 * 2;
    const size_t ybase = (size_t)im * 6 * bs;
    for (int q = lane; q < 64; q += 32) {
        int qi = q >> 3, qj = q & 7;
        size_t p = ybase + (size_t)((h0 >> 1) + qi) * (256 * 2) + (size_t)((w0 >> 1) + qj) * 2;
        int p00 = (2 * qi) * 16 + 2 * qj;
        float a, b, c, d;
        a = Tlh[p00]; b = Tlh[p00 + 1]; c = Tlh[p00 + 16]; d = Tlh[p00 + 17];
        yh0[p + 0 * bs]     = (a - d) * SQH;  yh0[p + 0 * bs + 1] = (b + c) * SQH; // 15
        yh0[p + 5 * bs]     = (a + d) * SQH;  yh0[p + 5 * bs + 1] = (b - c) * SQH; // 165
        a = Thh[p00]; b = Thh[p00 + 1]; c = Thh[p00 + 16]; d = Thh[p00 + 17];
        yh0[p + 1 * bs]     = (a - d) * SQH;  yh0[p + 1 * bs + 1] = (b + c) * SQH; // 45
        yh0[p + 4 * bs]     = (a + d) * SQH;  yh0[p + 4 * bs + 1] = (b - c) * SQH; // 135
        a = Thl[p00]; b = Thl[p00 + 1]; c = Thl[p00 + 16]; d = Thl[p00 + 17];
        yh0[p + 2 * bs]     = (a - d) * SQH;  yh0[p + 2 * bs + 1] = (b + c) * SQH; // 75
        yh0[p + 3 * bs]     = (a + d) * SQH;  yh0[p + 3 * bs + 1] = (b - c) * SQH; // 105
    }
}

// ---------------------------------------------------------------------------
// _dfilt along W:  ya[i]=sum_s ha[9-s]*X[4i-8+2s], yb[i]=sum_s hb[9-s]*X[4i-7+2s]
// lowpass (h0b,h0a): out[2i]=ya, out[2i+1]=yb ; highpass (h1b,h1a): swapped.
// ---------------------------------------------------------------------------
__global__ __launch_bounds__(256) void dfilt_w(
    const float* __restrict__ X, float* __restrict__ outLo, float* __restrict__ outHi,
    int rows, int r,
    const float* __restrict__ h0a, const float* __restrict__ h0b,
    const float* __restrict__ h1a, const float* __restrict__ h1b)
{
    int q = r >> 2;
    int tid = blockIdx.x * 256 + threadIdx.x;
    int total = 64 * rows * q;
    if (tid >= total) return;
    int i  = tid % q;
    int yy = (tid / q) % rows;
    int im = tid / (q * rows);
    const float* xr = X + ((size_t)im * rows + yy) * r;
    float xv[20];
    #pragma unroll
    for (int t = 0; t < 20; ++t) xv[t] = xr[refl(4 * i - 8 + t, r)];
    float ya0 = 0, yb0 = 0, ya1 = 0, yb1 = 0;
    #pragma unroll
    for (int s = 0; s < 10; ++s) {
        float xe = xv[2 * s], xo = xv[2 * s + 1];
        ya0 += h0b[9 - s] * xe;  yb0 += h0a[9 - s] * xo;
        ya1 += h1b[9 - s] * xe;  yb1 += h1a[9 - s] * xo;
    }
    float* lr = outLo + ((size_t)im * rows + yy) * (r >> 1);
    float* hr = outHi + ((size_t)im * rows + yy) * (r >> 1);
    lr[2 * i] = ya0;  lr[2 * i + 1] = yb0;    // lowpass: first=ya(h0b)
    hr[2 * i] = yb1;  hr[2 * i + 1] = ya1;    // highpass: first=yb(h1a)
}

// ---------------------------------------------------------------------------
// _dfilt along H for lo/hi + ll output + fused q2c -> yh[level].
// One thread per output quad (qi,qj) per image.
// ---------------------------------------------------------------------------
__global__ __launch_bounds__(256) void dfilt_h_q2c(
    const float* __restrict__ lo, const float* __restrict__ hi,
    float* __restrict__ ll, float* __restrict__ yh,
    int r, int wc,
    const float* __restrict__ h0a, const float* __restrict__ h0b,
    const float* __restrict__ h1a, const float* __restrict__ h1b)
{
    int W2 = wc >> 1, H2 = r >> 2;
    int tid = blockIdx.x * 256 + threadIdx.x;
    int total = 64 * H2 * W2;
    if (tid >= total) return;
    int qj = tid % W2;
    int qi = (tid / W2) % H2;
    int im = tid / (W2 * H2);

    float f0a[10], f0b[10], f1a[10], f1b[10];
    #pragma unroll
    for (int s = 0; s < 10; ++s) { f0a[s] = h0a[s]; f0b[s] = h0b[s]; f1a[s] = h1a[s]; f1b[s] = h1b[s]; }

    float llv[2][2];          // [row parity][col]
    float band[3][2][2];      // [lh,hl,hh][row parity][col]
    #pragma unroll
    for (int cc = 0; cc < 2; ++cc) {
        int c = 2 * qj + cc;
        const float* lc = lo + (size_t)im * r * wc + c;
        const float* hc = hi + (size_t)im * r * wc + c;
        float lv[20], hv[20];
        #pragma unroll
        for (int t = 0; t < 20; ++t) {
            int rr = refl(4 * qi - 8 + t, r);
            lv[t] = lc[(size_t)rr * wc];
            hv[t] = hc[(size_t)rr * wc];
        }
        float ya_l0 = 0, yb_l0 = 0, ya_l1 = 0, yb_l1 = 0;
        float ya_h0 = 0, yb_h0 = 0, ya_h1 = 0, yb_h1 = 0;
        #pragma unroll
        for (int s = 0; s < 10; ++s) {
            float e = lv[2 * s], o = lv[2 * s + 1];
            float eh = hv[2 * s], oh = hv[2 * s + 1];
            ya_l0 += f0b[9 - s] * e;   yb_l0 += f0a[9 - s] * o;
            ya_l1 += f1b[9 - s] * e;   yb_l1 += f1a[9 - s] * o;
            ya_h0 += f0b[9 - s] * eh;  yb_h0 += f0a[9 - s] * oh;
            ya_h1 += f1b[9 - s] * eh;  yb_h1 += f1a[9 - s] * oh;
        }
        llv[0][cc] = ya_l0;  llv[1][cc] = yb_l0;       // ll: lowpass
        band[0][0][cc] = yb_l1; band[0][1][cc] = ya_l1; // lh: highpass (swapped)
        band[1][0][cc] = ya_h0; band[1][1][cc] = yb_h0; // hl: lowpass
        band[2][0][cc] = yb_h1; band[2][1][cc] = ya_h1; // hh: highpass
    }

    float* llp = ll + (size_t)im * (r >> 1) * wc;
    llp[(size_t)(2 * qi) * wc + 2 * qj]         = llv[0][0];
    llp[(size_t)(2 * qi) * wc + 2 * qj + 1]     = llv[0][1];
    llp[(size_t)(2 * qi + 1) * wc + 2 * qj]     = llv[1][0];
    llp[(size_t)(2 * qi + 1) * wc + 2 * qj + 1] = llv[1][1];

    const size_t bs = (size_t)H2 * W2 * 2;
    size_t p = (size_t)im * 6 * bs + (size_t)qi * (W2 * 2) + (size_t)qj * 2;
    // band order: [lh.z1, hh.z1, hl.z1, hl.z2, hh.z2, lh.z2]
    const int z1idx[3] = {0, 2, 1};   // lh->0, hl->2, hh->1
    const int z2idx[3] = {5, 3, 4};   // lh->5, hl->3, hh->4
    #pragma unroll
    for (int bnd = 0; bnd < 3; ++bnd) {
        float a = band[bnd][0][0], b = band[bnd][0][1];
        float c = band[bnd][1][0], d = band[bnd][1][1];
        yh[p + (size_t)z1idx[bnd] * bs]     = (a - d) * SQH;
        yh[p + (size_t)z1idx[bnd] * bs + 1] = (b + c) * SQH;
        yh[p + (size_t)z2idx[bnd] * bs]     = (a + d) * SQH;
        yh[p + (size_t)z2idx[bnd] * bs + 1] = (b - c) * SQH;
    }
}

extern "C" void kernel_launch(void* const* d_in, const int* in_sizes, int n_in,
                              void* d_out, int out_size, void* d_ws, size_t ws_size,
                              hipStream_t stream) {
    (void)in_sizes; (void)n_in; (void)out_size; (void)ws_size;
    const float* x   = (const float*)d_in[0];
    const float* h0o = (const float*)d_in[1];
    const float* h1o = (const float*)d_in[2];
    const float* h0a = (const float*)d_in[3];
    const float* h0b = (const float*)d_in[4];
    const float* h1a = (const float*)d_in[5];
    const float* h1b = (const float*)d_in[6];
    float* out = (float*)d_out;
    float* ws  = (float*)d_ws;

    const size_t NI = 64;                      // 4 batch * 16 channels
    const size_t IMG = 512 * 512;
    float* lo1 = ws;                           // [0, 16.78M)
    float* hi1 = ws + NI * IMG;                // [16.78M, 33.55M)
    float* ll1 = ws + 2 * NI * IMG;            // [33.55M, 50.33M)

    // output segments: ll | yh0 | yh1 | yh2
    float* o_ll  = out;                        // 64*128*128      = 1,048,576
    float* o_yh0 = out + 1048576;              // 64*6*256*256*2  = 50,331,648
    float* o_yh1 = out + 51380224;             // 64*6*128*128*2  = 12,582,912
    float* o_yh2 = out + 63963136;             // 64*6*64*64*2    =  3,145,728

    // ---- Level 1 ----
    rowfilt_l1<<<dim3(512, 64), 256, 0, stream>>>(x, lo1, hi1, h0o, h1o);
    colfilt_l1_wmma<<<dim3(32, 32, 64), 32, 0, stream>>>(lo1, hi1, h0o, h1o, ll1, o_yh0);

    // ---- Level 2 ---- (lo1/hi1 regions now free)
    float* lo2 = ws;                           // 64*512*256 = 8,388,608
    float* hi2 = ws + NI * 512 * 256;
    {
        int total = 64 * 512 * 128;
        dfilt_w<<<(total + 255) / 256, 256, 0, stream>>>(ll1, lo2, hi2, 512, 512,
                                                         h0a, h0b, h1a, h1b);
    }
    float* ll2 = ws + NI * IMG;                // reuse hi1 region; 64*256*256
    {
        int total = 64 * 128 * 128;
        dfilt_h_q2c<<<(total + 255) / 256, 256, 0, stream>>>(lo2, hi2, ll2, o_yh1,
                                                             512, 256, h0a, h0b, h1a, h1b);
    }

    // ---- Level 3 ---- (lo2/hi2 regions now free)
    float* lo3 = ws;                           // 64*256*128 = 2,097,152
    float* hi3 = ws + NI * 256 * 128;
    {
        int total = 64 * 256 * 64;
        dfilt_w<<<(total + 255) / 256, 256, 0, stream>>>(ll2, lo3, hi3, 256, 256,
                                                         h0a, h0b, h1a, h1b);
    }
    {
        int total = 64 * 64 * 64;
        dfilt_h_q2c<<<(total + 255) / 256, 256, 0, stream>>>(lo3, hi3, o_ll, o_yh2,
                                                             256, 128, h0a, h0b, h1a, h1b);
    }
}